// BottomPool_61452392071362
// MI455X (gfx1250) — compile-verified
//
#include <hip/hip_runtime.h>
#include <hip/hip_bf16.h>
#include <math.h>

// BottomPool: out[b,c,h,w] = max_{h' <= h} x[b,c,h',w]
// Shape (B=16, C=256, H=128, W=128) f32.
// Memory-bound streaming scan: one thread owns one (b,c, 4-wide w group)
// column and walks H linearly with a running max. b128 non-temporal
// loads/stores, 16-deep load pipeline per wave for HBM latency hiding.

typedef float v4f __attribute__((ext_vector_type(4)));

#define H_DIM   128
#define W_DIM   128
#define W_VEC   4                    // float4 per thread
#define W4      (W_DIM / W_VEC)      // 32 vec-columns per row
#define COL_STRIDE_V4 (W_DIM / W_VEC) // v4f elements per H step (= 32)
#define UNROLL  16

__global__ __launch_bounds__(256) void bottom_pool_cummax_kernel(
    const float* __restrict__ x, float* __restrict__ out) {
  const unsigned tid = blockIdx.x * blockDim.x + threadIdx.x;
  // tid -> (bc, w4): lanes consecutive in w4 => coalesced 512B per wave access
  const unsigned w4 = tid & (W4 - 1);
  const unsigned bc = tid >> 5;

  const size_t base = (size_t)bc * (H_DIM * W_DIM) + (size_t)w4 * W_VEC;
  const v4f* __restrict__ src = (const v4f*)(x + base);
  v4f* __restrict__ dst = (v4f*)(out + base);

  v4f m;
  m.x = -INFINITY; m.y = -INFINITY; m.z = -INFINITY; m.w = -INFINITY;

  // 128 rows, pipelined in groups of 16: issue all 16 independent b128
  // NT loads first (8KB in flight / wave), then the dependent max chain
  // + NT stores.
  for (int h0 = 0; h0 < H_DIM; h0 += UNROLL) {
    v4f v[UNROLL];
#pragma unroll
    for (int i = 0; i < UNROLL; ++i) {
      v[i] = __builtin_nontemporal_load(src + (size_t)(h0 + i) * COL_STRIDE_V4);
    }
#pragma unroll
    for (int i = 0; i < UNROLL; ++i) {
      m.x = fmaxf(m.x, v[i].x);
      m.y = fmaxf(m.y, v[i].y);
      m.z = fmaxf(m.z, v[i].z);
      m.w = fmaxf(m.w, v[i].w);
      __builtin_nontemporal_store(m, dst + (size_t)(h0 + i) * COL_STRIDE_V4);
    }
  }
}

extern "C" void kernel_launch(void* const* d_in, const int* in_sizes, int n_in,
                              void* d_out, int out_size, void* d_ws, size_t ws_size,
                              hipStream_t stream) {
  (void)in_sizes; (void)n_in; (void)out_size; (void)d_ws; (void)ws_size;
  const float* x = (const float*)d_in[0];
  float* out = (float*)d_out;

  // Total threads: B*C * (W/4) = 16*256 * 32 = 131072
  const int total_threads = 16 * 256 * W4;
  const int block = 256;                   // 8 wave32s per workgroup
  const int grid = total_threads / block;  // 512 blocks

  bottom_pool_cummax_kernel<<<grid, block, 0, stream>>>(x, out);
}